// _FFParams_36601711296835
// MI455X (gfx1250) — compile-verified
//
#include <hip/hip_runtime.h>

// CDNA5 / gfx1250: wave32, WMMA f32 16x16x4 for full-precision Hadamard stages.
//
// Per row (L=16384):  out = (1/L) * H * (G .* P(H * (B .* x)))
// H_16384 = H4 (bits 12-13) (x) H16 (bits 8-11) (x) H16 (4-7) (x) H16 (0-3)
// Each H16 stage: 64 independent 16x16 tiles, D = H16 * X via 4 chained
// v_wmma_f32_16x16x4_f32 (K=16 in four K-slices). H16[m][k] = (-1)^popc(m&k).

typedef __attribute__((ext_vector_type(2))) float v2f;
typedef __attribute__((ext_vector_type(8))) float v8f;

#define LROW      16384
#define NTHREADS  512
#define NWAVES    (NTHREADS / 32)

__device__ __forceinline__ float hsign(int m, int k) {
    return (__popc(m & k) & 1) ? -1.0f : 1.0f;
}

// Element address for a 16-point transform stage of stride S:
// groups g in [0,1024), in-group index k in [0,16).
template <int S>
__device__ __forceinline__ int stage_addr(int g, int k) {
    return (g / S) * (S * 16) + (g % S) + k * S;
}

// Apply H16 along stride-S digit to all 16384 elements in LDS.
template <int S>
__device__ __forceinline__ void h16_stage(float* __restrict__ buf,
                                          int wave, int lm, int kh,
                                          const v2f A[4]) {
    #pragma unroll
    for (int it = 0; it < 64 / NWAVES; ++it) {  // constant trip count: no exec guard
        const int T = it * NWAVES + wave;
        const int gcol = T * 16 + lm;           // this lane's group (B/D column)
        v8f c = {0.f, 0.f, 0.f, 0.f, 0.f, 0.f, 0.f, 0.f};
        #pragma unroll
        for (int t = 0; t < 4; ++t) {
            const int k0 = 4 * t + kh;          // kh = 0 (lanes 0-15) or 2 (16-31)
            v2f b;
            b.x = buf[stage_addr<S>(gcol, k0)];
            b.y = buf[stage_addr<S>(gcol, k0 + 1)];
            c = __builtin_amdgcn_wmma_f32_16x16x4_f32(
                    false, A[t], false, b, (short)0, c, false, false);
        }
        const int mh = kh * 4;                  // 0 or 8: row offset of half-wave
        #pragma unroll
        for (int p = 0; p < 8; ++p) {
            buf[stage_addr<S>(gcol, p + mh)] = c[p];
        }
    }
}

// H4 along the top digit (stride 4096): two radix-2 levels on float4 vectors.
__device__ __forceinline__ void h4_stage(float4* __restrict__ s4p, int tid) {
    #pragma unroll
    for (int g4 = tid; g4 < 1024; g4 += NTHREADS) {   // 2 iterations, b128 LDS ops
        float4 a = s4p[g4];
        float4 b = s4p[g4 + 1024];
        float4 c = s4p[g4 + 2048];
        float4 d = s4p[g4 + 3072];
        float4 r0, r1, r2, r3;
        r0.x = (a.x + b.x) + (c.x + d.x);  r1.x = (a.x - b.x) + (c.x - d.x);
        r2.x = (a.x + b.x) - (c.x + d.x);  r3.x = (a.x - b.x) - (c.x - d.x);
        r0.y = (a.y + b.y) + (c.y + d.y);  r1.y = (a.y - b.y) + (c.y - d.y);
        r2.y = (a.y + b.y) - (c.y + d.y);  r3.y = (a.y - b.y) - (c.y - d.y);
        r0.z = (a.z + b.z) + (c.z + d.z);  r1.z = (a.z - b.z) + (c.z - d.z);
        r2.z = (a.z + b.z) - (c.z + d.z);  r3.z = (a.z - b.z) - (c.z - d.z);
        r0.w = (a.w + b.w) + (c.w + d.w);  r1.w = (a.w - b.w) + (c.w - d.w);
        r2.w = (a.w + b.w) - (c.w + d.w);  r3.w = (a.w - b.w) - (c.w - d.w);
        s4p[g4]        = r0;
        s4p[g4 + 1024] = r1;
        s4p[g4 + 2048] = r2;
        s4p[g4 + 3072] = r3;
    }
}

__global__ void __launch_bounds__(NTHREADS)
srht_row_kernel(const float* __restrict__ x, const float* __restrict__ Bv,
                const float* __restrict__ G, const int* __restrict__ Pi,
                float* __restrict__ out) {
    __shared__ float buf[LROW];                 // 64 KB of the 320 KB WGP LDS

    const int tid  = threadIdx.x;
    const int lane = tid & 31;
    const int wave = (tid >> 5) & (NWAVES - 1); // bounded range: helps codegen
    const int lm   = lane & 15;
    const int kh   = (lane & 16) ? 2 : 0;

    // Constant A operand: H16 K-slices in the ISA 16x4-f32 A layout.
    // Pin the 8 sign values in VGPRs so the compiler does not rematerialize
    // the popcount/cndmask chains in front of every WMMA.
    float s0 = hsign(lm, 0 + kh),  s1 = hsign(lm, 1 + kh);
    float s2 = hsign(lm, 4 + kh),  s3 = hsign(lm, 5 + kh);
    float s4 = hsign(lm, 8 + kh),  s5 = hsign(lm, 9 + kh);
    float s6 = hsign(lm, 12 + kh), s7 = hsign(lm, 13 + kh);
    asm("" : "+v"(s0), "+v"(s1), "+v"(s2), "+v"(s3),
             "+v"(s4), "+v"(s5), "+v"(s6), "+v"(s7));
    v2f A[4];
    A[0].x = s0; A[0].y = s1;
    A[1].x = s2; A[1].y = s3;
    A[2].x = s4; A[2].y = s5;
    A[3].x = s6; A[3].y = s7;

    const size_t rowoff = (size_t)blockIdx.x * LROW;
    const float4* xr = (const float4*)(x + rowoff);
    const float4* bv = (const float4*)Bv;
    const float4* g4 = (const float4*)G;
    const int4*   p4 = (const int4*)Pi;
    float4* o4  = (float4*)(out + rowoff);
    float4* s4p = (float4*)buf;

    // Load row, fold in the +/-1 diagonal B.  b128 loads, fully coalesced.
    #pragma unroll
    for (int i = tid; i < LROW / 4; i += NTHREADS) {
        float4 v = xr[i];
        float4 b = bv[i];
        v.x *= b.x; v.y *= b.y; v.z *= b.z; v.w *= b.w;
        s4p[i] = v;
    }
    __syncthreads();

    // ---- FWHT #1 ----
    h16_stage<1>(buf, wave, lm, kh, A);   __syncthreads();
    h16_stage<16>(buf, wave, lm, kh, A);  __syncthreads();
    h16_stage<256>(buf, wave, lm, kh, A); __syncthreads();
    h4_stage(s4p, tid);                   __syncthreads();

    // ---- Permute + G diagonal; fold both 1/sqrt(L) scales in here ----
    // Pi/G read as b128; LDS gather is scalar (it's a true permutation);
    // scatter-back is b128.
    const float sc = 1.0f / 16384.0f;
    float4 st[8];
    #pragma unroll
    for (int i = 0; i < 8; ++i) {
        const int j4 = i * NTHREADS + tid;
        const int4   p = p4[j4];
        const float4 g = g4[j4];
        st[i].x = buf[p.x] * (g.x * sc);
        st[i].y = buf[p.y] * (g.y * sc);
        st[i].z = buf[p.z] * (g.z * sc);
        st[i].w = buf[p.w] * (g.w * sc);
    }
    __syncthreads();
    #pragma unroll
    for (int i = 0; i < 8; ++i) {
        s4p[i * NTHREADS + tid] = st[i];
    }
    __syncthreads();

    // ---- FWHT #2 ----
    h16_stage<1>(buf, wave, lm, kh, A);   __syncthreads();
    h16_stage<16>(buf, wave, lm, kh, A);  __syncthreads();
    h16_stage<256>(buf, wave, lm, kh, A); __syncthreads();
    h4_stage(s4p, tid);                   __syncthreads();

    // Pure streaming store (scale already folded in).
    #pragma unroll
    for (int i = tid; i < LROW / 4; i += NTHREADS) {
        o4[i] = s4p[i];
    }
}

extern "C" void kernel_launch(void* const* d_in, const int* in_sizes, int n_in,
                              void* d_out, int out_size, void* d_ws, size_t ws_size,
                              hipStream_t stream) {
    const float* x  = (const float*)d_in[0];   // (4096, 16384) fp32
    const float* B  = (const float*)d_in[1];   // (16384,) fp32 (+/-1)
    const float* G  = (const float*)d_in[2];   // (16384,) fp32
    const int*   Pi = (const int*)d_in[3];     // (16384,) int32 permutation
    float* out = (float*)d_out;

    const int rows = in_sizes[0] / LROW;       // 4096
    hipLaunchKernelGGL(srht_row_kernel, dim3(rows), dim3(NTHREADS), 0, stream,
                       x, B, G, Pi, out);
}